// DivisionIteration25_23407571764114
// MI455X (gfx1250) — compile-verified
//
#include <hip/hip_runtime.h>
#include <stdint.h>

// SNN bit-serial division iteration, collapsed to integer ops.
// Per 25-bit row (LSB-first, exact {0,1} f32 bits):
//   r = pack(R); d = pack(D);
//   ge = (r >= d);  rn = ge ? (r - d) & 0x1FFFFFF : r;
//   Q = ge;  R_next = unpack(rn)
// Memory-bound: 637 MB streamed -> ~27us floor at 23.3 TB/s.
// All traffic is b128; stores are non-temporal (no reuse, keep L2 clean).

constexpr int BITS = 25;

typedef float v4f __attribute__((ext_vector_type(4)));

__device__ __forceinline__ uint32_t bit_of(float x) {
    // x is exactly 0.0f or 1.0f; exponent-field trick avoids v_cvt:
    // 1.0f = 0x3F800000 -> (u >> 23) & 1 == 1 ; 0.0f -> 0.
    return (__float_as_uint(x) >> 23) & 1u;
}

__global__ __launch_bounds__(256) void snn_div_iter(
    const float* __restrict__ R, const float* __restrict__ D,
    float* __restrict__ Q, float* __restrict__ Rn, int nRows)
{
    const int t = blockIdx.x * blockDim.x + threadIdx.x;
    const long long row0 = (long long)t * 4;
    if (row0 >= nRows) return;

    if (row0 + 4 <= nRows) {
        // Fast path: 4 rows = 400B per operand = 25 aligned float4 loads.
        const v4f* __restrict__ R4 = reinterpret_cast<const v4f*>(R + row0 * BITS);
        const v4f* __restrict__ D4 = reinterpret_cast<const v4f*>(D + row0 * BITS);

        // gfx1250 prefetch path (global_prefetch_b8): warm the tile.
        __builtin_prefetch(R4, 0, 0);
        __builtin_prefetch(D4, 0, 0);

        uint32_t r[4] = {0u, 0u, 0u, 0u};
        uint32_t d[4] = {0u, 0u, 0u, 0u};

        #pragma unroll
        for (int v = 0; v < 25; ++v) {
            const v4f rv = R4[v];
            const v4f dv = D4[v];
            #pragma unroll
            for (int j = 0; j < 4; ++j) {
                const int flat = v * 4 + j;   // 0..99 flat float index
                const int e    = flat / 25;   // row within 4-row group
                const int bpos = flat % 25;   // bit position (LSB first)
                r[e] |= bit_of(rv[j]) << bpos;
                d[e] |= bit_of(dv[j]) << bpos;
            }
        }

        uint32_t rn[4];
        v4f qv;
        #pragma unroll
        for (int e = 0; e < 4; ++e) {
            const uint32_t diff = (r[e] - d[e]) & 0x1FFFFFFu;
            const bool ge = (r[e] >= d[e]);
            rn[e] = ge ? diff : r[e];
            qv[e] = ge ? 1.0f : 0.0f;
        }

        // Q: 4 consecutive floats, 16B aligned (row0 % 4 == 0).
        __builtin_nontemporal_store(qv, reinterpret_cast<v4f*>(Q + row0));

        // R_next: 100 floats = 25 aligned float4 NT stores.
        v4f* __restrict__ O4 = reinterpret_cast<v4f*>(Rn + row0 * BITS);
        #pragma unroll
        for (int v = 0; v < 25; ++v) {
            v4f ov;
            #pragma unroll
            for (int j = 0; j < 4; ++j) {
                const int flat = v * 4 + j;
                const int e    = flat / 25;
                const int bpos = flat % 25;
                ov[j] = ((rn[e] >> bpos) & 1u) ? 1.0f : 0.0f;
            }
            __builtin_nontemporal_store(ov, O4 + v);
        }
    } else {
        // Scalar tail (never taken for N = 2^21, kept for robustness).
        for (long long row = row0; row < nRows; ++row) {
            uint32_t r = 0u, d = 0u;
            const float* Rr = R + row * BITS;
            const float* Dr = D + row * BITS;
            for (int b = 0; b < BITS; ++b) {
                r |= bit_of(Rr[b]) << b;
                d |= bit_of(Dr[b]) << b;
            }
            const uint32_t diff = (r - d) & 0x1FFFFFFu;
            const bool ge = (r >= d);
            const uint32_t rn = ge ? diff : r;
            Q[row] = ge ? 1.0f : 0.0f;
            float* Or = Rn + row * BITS;
            for (int b = 0; b < BITS; ++b)
                Or[b] = ((rn >> b) & 1u) ? 1.0f : 0.0f;
        }
    }
}

extern "C" void kernel_launch(void* const* d_in, const int* in_sizes, int n_in,
                              void* d_out, int out_size, void* d_ws, size_t ws_size,
                              hipStream_t stream) {
    const float* R = (const float*)d_in[0];
    const float* D = (const float*)d_in[1];
    float* out = (float*)d_out;

    const int nRows = in_sizes[0] / BITS;   // 2,097,152
    float* Q  = out;            // [nRows]    (Q_bit, shape [N,1] flattened)
    float* Rn = out + nRows;    // [nRows*25] (R_next row-major)

    const int nGroups = (nRows + 3) / 4;    // 4 rows per thread
    const int block = 256;                  // 8 wave32s per block
    const int grid = (nGroups + block - 1) / block;

    hipLaunchKernelGGL(snn_div_iter, dim3(grid), dim3(block), 0, stream,
                       R, D, Q, Rn, nRows);
}